// AnovaKernel_47347719471416
// MI455X (gfx1250) — compile-verified
//
#include <hip/hip_runtime.h>

// ANOVA kernel, order 3:  out[b] = sum_e e3(x[b, :, e])
// x: (8192, 64, 64) fp32.  Memory-bound streaming reduction (134 MB @ 23.3 TB/s).
//
// Design:
//  - wave32: one wave per batch, lane owns e = {2*lane, 2*lane+1} (float2 per field row)
//  - double-buffered GLOBAL_LOAD_ASYNC_TO_LDS_B64 staging (ASYNCcnt pipelining)
//  - per-(b,e) DP scan in fp32 VALU (exact vs reference)
//  - final 64-lane embedding reduction via V_WMMA_F32_16X16X4_F32 with ones-B
//    (wave's 64 partials are exactly the 16x4 f32 A-matrix VGPR layout)

#define NF 64
#define NE 64
#define WPB 8            // waves per block (256 threads)
#define CH 8             // fields per async chunk
#define NCH (NF / CH)    // 8 chunks

typedef float v2f __attribute__((ext_vector_type(2)));
typedef float v8f __attribute__((ext_vector_type(8)));
// The async-load builtins take generic-vector int pointers (per hipcc diagnostic):
typedef int b64vec __attribute__((vector_size(8)));

#if __has_builtin(__builtin_amdgcn_global_load_async_to_lds_b64) && \
    __has_builtin(__builtin_amdgcn_s_wait_asynccnt)
#define HAVE_ASYNC 1
#else
#define HAVE_ASYNC 0
#endif

#if __has_builtin(__builtin_amdgcn_wmma_f32_16x16x4_f32)
#define HAVE_WMMA_F32 1
#else
#define HAVE_WMMA_F32 0
#endif

__global__ __launch_bounds__(WPB * 32)
void anova3_kernel(const float* __restrict__ x, float* __restrict__ out, int nbatch) {
  const int wave = threadIdx.x >> 5;
  const int lane = threadIdx.x & 31;
  const int b = blockIdx.x * WPB + wave;
  if (b >= nbatch) return;  // wave-uniform branch; EXEC stays all-ones inside wave

  const float* xb = x + (size_t)b * (NF * NE);

  // DP accumulators for the lane's two embedding channels
  float a1x = 0.f, a2x = 0.f, a3x = 0.f;
  float a1y = 0.f, a2y = 0.f, a3y = 0.f;

#if HAVE_ASYNC
  __shared__ __align__(16) float lds[WPB][2][CH * NE];  // 32 KB/block, double buffered

#define ISSUE_CHUNK(c, bufi)                                                   \
  do {                                                                         \
    _Pragma("unroll")                                                          \
    for (int i = 0; i < CH; ++i) {                                             \
      __builtin_amdgcn_global_load_async_to_lds_b64(                           \
          (__attribute__((address_space(1))) b64vec*)                          \
              (xb + ((c) * CH + i) * NE + 2 * lane),                           \
          (__attribute__((address_space(3))) b64vec*)                          \
              (&lds[wave][(bufi)][i * NE + 2 * lane]),                         \
          0, 0);                                                               \
    }                                                                          \
  } while (0)

  ISSUE_CHUNK(0, 0);
  for (int c = 0; c < NCH; ++c) {
    if (c + 1 < NCH) {
      ISSUE_CHUNK(c + 1, (c + 1) & 1);
      __builtin_amdgcn_s_wait_asynccnt(CH);  // chunk c has landed in LDS
    } else {
      __builtin_amdgcn_s_wait_asynccnt(0);
    }
    asm volatile("" ::: "memory");
    const float* buf = &lds[wave][c & 1][0];
#pragma unroll
    for (int i = 0; i < CH; ++i) {
      v2f v = *(const v2f*)(buf + i * NE + 2 * lane);
      a3x += a2x * v.x; a2x += a1x * v.x; a1x += v.x;
      a3y += a2y * v.y; a2y += a1y * v.y; a1y += v.y;
    }
  }
#undef ISSUE_CHUNK
#else
  // Fallback: direct coalesced global_load_b64 stream
#pragma unroll 8
  for (int f = 0; f < NF; ++f) {
    v2f v = *(const v2f*)(xb + f * NE + 2 * lane);
    a3x += a2x * v.x; a2x += a1x * v.x; a1x += v.x;
    a3y += a2y * v.y; a2y += a1y * v.y; a1y += v.y;
  }
#endif

  // ---- reduce the wave's 64 fp32 partials (all e of this batch) to one float
  float total;
#if HAVE_WMMA_F32
  // 16x4 f32 A-matrix layout == exactly our 2 partials per lane across 32 lanes.
  // D = A x ones(4x16): D[m][n] = row_sum(A[m]).  Sum 8 D-VGPRs per lane gives
  // half the grand total; one xor-16 shuffle-add completes it.
  v2f A;  A.x = a3x; A.y = a3y;
  v2f Bm; Bm.x = 1.0f; Bm.y = 1.0f;
  v8f Cm = {};
  v8f D = __builtin_amdgcn_wmma_f32_16x16x4_f32(
      /*neg_a=*/false, A, /*neg_b=*/false, Bm,
      /*c_mod=*/(short)0, Cm, /*reuse_a=*/false, /*reuse_b=*/false);
  float t = D[0] + D[1] + D[2] + D[3] + D[4] + D[5] + D[6] + D[7];
  total = t + __shfl_xor(t, 16, 32);
#else
  float s = a3x + a3y;
#pragma unroll
  for (int m = 16; m >= 1; m >>= 1) s += __shfl_xor(s, m, 32);
  total = s;
#endif

  if (lane == 0) out[b] = total;
}

extern "C" void kernel_launch(void* const* d_in, const int* in_sizes, int n_in,
                              void* d_out, int out_size, void* d_ws, size_t ws_size,
                              hipStream_t stream) {
  (void)n_in; (void)out_size; (void)d_ws; (void)ws_size;
  const float* x = (const float*)d_in[0];
  float* out = (float*)d_out;
  const int nbatch = in_sizes[0] / (NF * NE);  // 8192
  const int blocks = (nbatch + WPB - 1) / WPB; // 1024
  anova3_kernel<<<blocks, WPB * 32, 0, stream>>>(x, out, nbatch);
}